// my_model4_86053964743257
// MI455X (gfx1250) — compile-verified
//
#include <hip/hip_runtime.h>
#include <math.h>

// ---------------- problem constants ----------------
static constexpr int B_  = 4;
static constexpr int L_  = 1024;
static constexpr int H_  = 768;
static constexpr int M_  = 96;
static constexpr int E_  = 32;
static constexpr int HD_ = 12;   // HEADS
static constexpr int R_  = 97;
static constexpr int BS_ = 64;
static constexpr int P_  = E_ * (E_ - 1);        // 992
static constexpr int BP_ = B_ * P_;              // 3968
static constexpr int KB_ = H_ / BS_;             // 12 blocks
static constexpr int F_  = KB_ * BS_ * BS_;      // 49152 features

typedef __attribute__((ext_vector_type(2))) float v2f;
typedef __attribute__((ext_vector_type(8))) float v8f;

__device__ __forceinline__ v8f wmma_f32(v2f a, v2f b, v8f c) {
  // V_WMMA_F32_16X16X4_F32 : D = A(16x4) x B(4x16) + C(16x16), full f32
  return __builtin_amdgcn_wmma_f32_16x16x4_f32(
      /*neg_a=*/false, a, /*neg_b=*/false, b,
      /*c_mod=*/(short)0, c, /*reuse_a=*/false, /*reuse_b=*/false);
}

// ---------------- kernel 1: token index extraction (mention_map is one-hot) ----
__global__ __launch_bounds__(128)
void tok_kernel(const float* __restrict__ mention_map, int* __restrict__ tok) {
  int i = blockIdx.x * 128 + threadIdx.x;       // b*M + m
  if (i >= B_ * M_) return;
  const float* row = mention_map + (long)i * L_;
  int idx = 0;
  for (int l = 0; l < L_; ++l)
    if (row[l] > 0.5f) idx = l;
  tok[i] = idx;
}

// ---------------- kernel 2: entity = log(sum_m w * exp(context[tok_m])) -------
__global__ __launch_bounds__(256)
void entity_kernel(const float* __restrict__ context,
                   const float* __restrict__ entity_map,
                   const int* __restrict__ tok,
                   float* __restrict__ entity) {
  int be = blockIdx.x;                 // b*E + e
  int b  = be / E_;
  const float* em = entity_map + (long)be * M_;
  __shared__ float w[M_];
  __shared__ int   tk[M_];
  if (threadIdx.x < M_) {
    w[threadIdx.x]  = em[threadIdx.x];
    tk[threadIdx.x] = tok[b * M_ + threadIdx.x];
  }
  __syncthreads();
  for (int hh = threadIdx.x; hh < H_; hh += 256) {
    float acc = 0.f, rs = 0.f;
    for (int m = 0; m < M_; ++m) {
      float ww = w[m];
      rs += ww;
      if (ww != 0.f)
        acc += ww * expf(context[((long)b * L_ + tk[m]) * H_ + hh]);
    }
    entity[(long)be * H_ + hh] = (rs == 0.f) ? 0.f : logf(acc);
  }
}

// ---------------- kernel 3: ea[b,hd,e,:] = (sum_m w*att[tok_m,:]) / sum(w) ----
__global__ __launch_bounds__(256)
void ea_kernel(const float* __restrict__ attention,
               const float* __restrict__ entity_map,
               const int* __restrict__ tok,
               float* __restrict__ ea) {
  int id = blockIdx.x;                 // ((b*HD + hd)*E + e)
  int e = id % E_;
  int t2 = id / E_;
  int hd = t2 % HD_;
  int b  = t2 / HD_;
  const float* em = entity_map + ((long)b * E_ + e) * M_;
  __shared__ float w[M_];
  __shared__ int   tk[M_];
  if (threadIdx.x < M_) {
    w[threadIdx.x]  = em[threadIdx.x];
    tk[threadIdx.x] = tok[b * M_ + threadIdx.x];
  }
  __syncthreads();
  float denom = 0.f;
  for (int m = 0; m < M_; ++m) denom += w[m];
  float inv = 1.f / (denom + 1e-20f);
  const float* att = attention + ((long)b * HD_ + hd) * (long)L_ * L_;
  for (int l = threadIdx.x; l < L_; l += 256) {
    float s = 0.f;
    for (int m = 0; m < M_; ++m) {
      float ww = w[m];
      if (ww != 0.f) s += ww * att[(long)tk[m] * L_ + l];
    }
    ea[(long)id * L_ + l] = s * inv;
  }
}

// ---------------- kernel 4: ca[b,p,:] = norm(sum_hd ea[h]*ea[t]) * mask -------
__global__ __launch_bounds__(256)
void ca_kernel(const float* __restrict__ ea,
               const int* __restrict__ hts,
               float* __restrict__ ca) {
  int bp = blockIdx.x;                 // b*P + p
  int b  = bp / P_;
  int he = hts[bp * 2 + 0];
  int te = hts[bp * 2 + 1];
  float mask = (he + te != 0) ? 1.f : 0.f;
  const float* eb = ea + (long)b * HD_ * E_ * L_;
  float* crow = ca + (long)bp * L_;
  float lsum = 0.f;
  for (int l = threadIdx.x; l < L_; l += 256) {
    float s = 0.f;
    for (int hd = 0; hd < HD_; ++hd) {
      const float* e1 = eb + ((long)hd * E_ + he) * L_;
      const float* e2 = eb + ((long)hd * E_ + te) * L_;
      s += e1[l] * e2[l];
    }
    crow[l] = s;
    lsum += s;
  }
  __shared__ float red[256];
  red[threadIdx.x] = lsum;
  __syncthreads();
  for (int off = 128; off > 0; off >>= 1) {
    if (threadIdx.x < off) red[threadIdx.x] += red[threadIdx.x + off];
    __syncthreads();
  }
  float scale = mask / (red[0] + 1e-20f);
  for (int l = threadIdx.x; l < L_; l += 256) crow[l] *= scale;
}

// ---------------- kernel 5: gather h/t entity rows with mask ------------------
__global__ __launch_bounds__(256)
void gather_kernel(const float* __restrict__ entity,
                   const int* __restrict__ hts,
                   float* __restrict__ hg, float* __restrict__ tg) {
  long idx = (long)blockIdx.x * 256 + threadIdx.x;   // over BP*H
  if (idx >= (long)BP_ * H_) return;
  int  k  = (int)(idx % H_);
  long bp = idx / H_;
  int  b  = (int)(bp / P_);
  int he = hts[bp * 2 + 0];
  int te = hts[bp * 2 + 1];
  float mask = (he + te != 0) ? 1.f : 0.f;
  hg[idx] = entity[((long)b * E_ + he) * H_ + k] * mask;
  tg[idx] = entity[((long)b * E_ + te) * H_ + k] * mask;
}

// ---------------- WMMA f32 GEMM: D = act(A1@B1 [+ A2@B2] + bias1 + bias2) ----
// Block tile 64x64, 8 waves (256 threads); each wave computes a 16x32 strip
// (two 16x16 WMMA tiles sharing one A operand).
// transB: B[k][n] = W[n*ldb + k] (i.e. W is row-major (N,K), we use W^T).
__global__ __launch_bounds__(256)
void wmma_gemm_kernel(const float* __restrict__ A1, const float* __restrict__ B1,
                      const float* __restrict__ A2, const float* __restrict__ B2,
                      const float* __restrict__ bias1, const float* __restrict__ bias2,
                      float* __restrict__ D,
                      int Mrows, int Kd,
                      int lda, int ldb, int ldd,
                      int transB, int act,
                      long sA, long sB, long sD) {
  __shared__ float As[64 * 20];   // [m][k], row stride 20 (pad, 16B-aligned rows)
  __shared__ float Bs[16 * 72];   // [k][n], row stride 72 (2*72 % 64 == 16 -> halves hit disjoint banks)

  int tid  = threadIdx.x;
  int lane = tid & 31;
  int wave = tid >> 5;            // 0..7
  int wm   = wave & 3;            // M tile within block (0..3)
  int wn   = wave >> 2;           // N strip within block (0..1), 32 cols each
  int half = lane >> 4;
  int l16  = lane & 15;
  int bz   = blockIdx.z;
  int m0   = blockIdx.y * 64;
  int n0   = blockIdx.x * 64;

  v8f c0 = {0.f, 0.f, 0.f, 0.f, 0.f, 0.f, 0.f, 0.f};
  v8f c1 = {0.f, 0.f, 0.f, 0.f, 0.f, 0.f, 0.f, 0.f};

  for (int pass = 0; pass < 2; ++pass) {
    const float* Ap;
    const float* Bp;
    if (pass == 0) { Ap = A1 + (long)bz * sA; Bp = B1 + (long)bz * sB; }
    else           { if (A2 == nullptr) break; Ap = A2; Bp = B2; }

    for (int kk = 0; kk < Kd; kk += 16) {
      // stage A tile (64x16), coalesced float4
      {
        int row = tid >> 2;               // 0..63
        int kq  = (tid & 3) * 4;          // 0,4,8,12
        int gr  = m0 + row;
        float4 v = make_float4(0.f, 0.f, 0.f, 0.f);
        if (gr < Mrows) v = *(const float4*)(Ap + (long)gr * lda + kk + kq);
        *(float4*)(&As[row * 20 + kq]) = v;
      }
      // stage B tile (16x64)
      if (!transB) {
        int row = tid >> 4;               // k 0..15
        int nn  = (tid & 15) * 4;         // n 0..60
        float4 v = *(const float4*)(Bp + (long)(kk + row) * ldb + n0 + nn);
        *(float4*)(&Bs[row * 72 + nn]) = v;
      } else {
        int nn = tid >> 2;                // n 0..63
        int k2 = (tid & 3) * 4;           // k 0,4,8,12
        float4 v = *(const float4*)(Bp + (long)(n0 + nn) * ldb + kk + k2);
        Bs[(k2 + 0) * 72 + nn] = v.x;
        Bs[(k2 + 1) * 72 + nn] = v.y;
        Bs[(k2 + 2) * 72 + nn] = v.z;
        Bs[(k2 + 3) * 72 + nn] = v.w;
      }
      __syncthreads();
#pragma unroll
      for (int ks = 0; ks < 16; ks += 4) {
        int ko = ks + 2 * half;
        float2 av = *(const float2*)(&As[(wm * 16 + l16) * 20 + ko]);
        v2f a; a.x = av.x; a.y = av.y;
        int cb = wn * 32 + l16;
        v2f b0, b1;
        b0.x = Bs[ko * 72 + cb];
        b0.y = Bs[(ko + 1) * 72 + cb];
        b1.x = Bs[ko * 72 + cb + 16];
        b1.y = Bs[(ko + 1) * 72 + cb + 16];
        c0 = wmma_f32(a, b0, c0);
        c1 = wmma_f32(a, b1, c1);
      }
      __syncthreads();
    }
  }

  int col0 = n0 + wn * 32 + l16;          // always < N (N multiple of 64)
  int col1 = col0 + 16;
  float bia0 = 0.f, bia1 = 0.f;
  if (bias1) { bia0 += bias1[col0]; bia1 += bias1[col1]; }
  if (bias2) { bia0 += bias2[col0]; bia1 += bias2[col1]; }
  float* Dp = D + (long)bz * sD;
#pragma unroll
  for (int i = 0; i < 8; ++i) {
    int rowm = m0 + wm * 16 + i + 8 * half;
    if (rowm < Mrows) {
      float v0 = c0[i] + bia0;
      float v1 = c1[i] + bia1;
      if (act) { v0 = tanhf(v0); v1 = tanhf(v1); }
      Dp[(long)rowm * ldd + col0] = v0;
      Dp[(long)rowm * ldd + col1] = v1;
    }
  }
}

// ---------------- kernel 7: fused block-bilinear via WMMA ---------------------
// pred[bp, r] = sum_{k,d,e} h[bp,k*64+d] * Wc[r,k*4096+d*64+e] * t[bp,k*64+e] + bc[r]
// == Z @ Wc^T with Z built on the fly. One block = 64 pairs, 7 waves (224 thr);
// wave w owns r-tile w (R=97 -> 7 tiles of 16). Each wave carries FOUR 16-pair
// accumulators so every Wc B-operand load feeds 4 WMMAs (4x less L2 traffic).
// h/t are staged per k-slice (64 pairs x 64 dims) -> 34 KB LDS.
__global__ __launch_bounds__(224)
void bilinear_kernel(const float* __restrict__ hq, const float* __restrict__ tq,
                     const float* __restrict__ Wc, const float* __restrict__ bc,
                     float* __restrict__ out) {
  __shared__ float hs[64 * 68];     // [pair][d], stride 68 -> bank = (4*p + d) % 64
  __shared__ float ts[64 * 68];

  int tid  = threadIdx.x;
  int lane = tid & 31;
  int wave = tid >> 5;              // 0..6 == r-tile index
  int half = lane >> 4;
  int l16  = lane & 15;
  long p0  = (long)blockIdx.x * 64; // pair-row base

  int rg = wave * 16 + l16;         // 0..111; rows >= 97 clamped (discarded at store)
  const float* wrow = Wc + (long)min(rg, R_ - 1) * F_;

  v8f acc0 = {0.f, 0.f, 0.f, 0.f, 0.f, 0.f, 0.f, 0.f};
  v8f acc1 = acc0, acc2 = acc0, acc3 = acc0;

  for (int k = 0; k < KB_; ++k) {
    // stage the k-th 64-wide slice of h/t for 64 pairs (each element read once)
    for (int idx = tid; idx < 64 * 64; idx += 224) {
      int p = idx >> 6, d = idx & 63;
      hs[p * 68 + d] = hq[(p0 + p) * H_ + k * 64 + d];
      ts[p * 68 + d] = tq[(p0 + p) * H_ + k * 64 + d];
    }
    __syncthreads();

    long fk = (long)k * 4096;
    for (int d = 0; d < 64; ++d) {
      float hv0 = hs[(l16)*68 + d];
      float hv1 = hs[(16 + l16) * 68 + d];
      float hv2 = hs[(32 + l16) * 68 + d];
      float hv3 = hs[(48 + l16) * 68 + d];
      long fb = fk + (long)d * 64;
      if (fb + 192 < F_)
        __builtin_prefetch(wrow + fb + 192, 0, 0);   // prefetch Wc stream ahead
#pragma unroll
      for (int e0 = 0; e0 < 64; e0 += 4) {
        int eo = e0 + 2 * half;
        float2 w = *(const float2*)(wrow + fb + eo);
        v2f bv; bv.x = w.x; bv.y = w.y;
        float2 t0 = *(const float2*)(&ts[(l16)*68 + eo]);
        float2 t1 = *(const float2*)(&ts[(16 + l16) * 68 + eo]);
        float2 t2 = *(const float2*)(&ts[(32 + l16) * 68 + eo]);
        float2 t3 = *(const float2*)(&ts[(48 + l16) * 68 + eo]);
        v2f a0; a0.x = hv0 * t0.x; a0.y = hv0 * t0.y;
        v2f a1; a1.x = hv1 * t1.x; a1.y = hv1 * t1.y;
        v2f a2; a2.x = hv2 * t2.x; a2.y = hv2 * t2.y;
        v2f a3; a3.x = hv3 * t3.x; a3.y = hv3 * t3.y;
        acc0 = wmma_f32(a0, bv, acc0);
        acc1 = wmma_f32(a1, bv, acc1);
        acc2 = wmma_f32(a2, bv, acc2);
        acc3 = wmma_f32(a3, bv, acc3);
      }
    }
    __syncthreads();
  }

  if (rg < R_) {
    float bias = bc[rg];
#pragma unroll
    for (int i = 0; i < 8; ++i) {
      int pr = i + 8 * half;        // pair within 16-tile (C-matrix M dim)
      out[(p0 + 0 * 16 + pr) * R_ + rg] = acc0[i] + bias;
      out[(p0 + 1 * 16 + pr) * R_ + rg] = acc1[i] + bias;
      out[(p0 + 2 * 16 + pr) * R_ + rg] = acc2[i] + bias;
      out[(p0 + 3 * 16 + pr) * R_ + rg] = acc3[i] + bias;
    }
  }
}

// ---------------- host launcher ----------------
extern "C" void kernel_launch(void* const* d_in, const int* in_sizes, int n_in,
                              void* d_out, int out_size, void* d_ws, size_t ws_size,
                              hipStream_t stream) {
  const float* context    = (const float*)d_in[0];
  const float* attention  = (const float*)d_in[1];
  const float* mention_mp = (const float*)d_in[2];
  const float* entity_map = (const float*)d_in[3];
  const int*   hts        = (const int*)  d_in[4];
  const float* Wh  = (const float*)d_in[5];
  const float* bh  = (const float*)d_in[6];
  const float* Wt  = (const float*)d_in[7];
  const float* bt  = (const float*)d_in[8];
  const float* Whc = (const float*)d_in[9];
  const float* bhc = (const float*)d_in[10];
  const float* Wtc = (const float*)d_in[11];
  const float* btc = (const float*)d_in[12];
  const float* Wc  = (const float*)d_in[13];
  const float* bc  = (const float*)d_in[14];
  float* out = (float*)d_out;

  char* ws = (char*)d_ws;
  size_t off = 0;
  auto take = [&](size_t bytes) -> void* {
    off = (off + 255) & ~size_t(255);
    void* p = ws + off;
    off += bytes;
    return p;
  };
  int*   tok    = (int*)  take((size_t)B_ * M_ * sizeof(int));
  float* entity = (float*)take((size_t)B_ * E_ * H_ * sizeof(float));
  float* ea     = (float*)take((size_t)B_ * HD_ * E_ * L_ * sizeof(float));
  float* ca     = (float*)take((size_t)BP_ * L_ * sizeof(float));
  float* hg     = (float*)take((size_t)BP_ * H_ * sizeof(float));
  float* tg     = (float*)take((size_t)BP_ * H_ * sizeof(float));
  float* ci     = (float*)take((size_t)BP_ * H_ * sizeof(float));
  float* hq     = (float*)take((size_t)BP_ * H_ * sizeof(float));
  float* tq     = (float*)take((size_t)BP_ * H_ * sizeof(float));
  (void)ws_size; (void)in_sizes; (void)n_in; (void)out_size;

  // 1) token indices
  tok_kernel<<<(B_ * M_ + 127) / 128, 128, 0, stream>>>(mention_mp, tok);
  // 2) entity log-sum-exp embeddings
  entity_kernel<<<B_ * E_, 256, 0, stream>>>(context, entity_map, tok, entity);
  // 3) entity attention rows
  ea_kernel<<<B_ * HD_ * E_, 256, 0, stream>>>(attention, entity_map, tok, ea);
  // 4) pair context weights (normalized, masked)
  ca_kernel<<<BP_, 256, 0, stream>>>(ea, hts, ca);
  // 5) context_info = ca @ context  (batched WMMA GEMM, M=992 N=768 K=1024)
  {
    dim3 g(H_ / 64, (P_ + 63) / 64, B_);
    wmma_gemm_kernel<<<g, 256, 0, stream>>>(
        ca, context, nullptr, nullptr, nullptr, nullptr, ci,
        P_, L_, L_, H_, H_, /*transB=*/0, /*act=*/0,
        (long)P_ * L_, (long)L_ * H_, (long)P_ * H_);
  }
  // 6) gather masked h/t rows
  {
    long n = (long)BP_ * H_;
    gather_kernel<<<(unsigned)((n + 255) / 256), 256, 0, stream>>>(entity, hts, hg, tg);
  }
  // 7) h = tanh(hg@Wh^T + ci@Whc^T + bh + bhc)   (dual-pass WMMA GEMM)
  {
    dim3 g(H_ / 64, BP_ / 64, 1);
    wmma_gemm_kernel<<<g, 256, 0, stream>>>(
        hg, Wh, ci, Whc, bh, bhc, hq,
        BP_, H_, H_, H_, H_, /*transB=*/1, /*act=*/1, 0, 0, 0);
    wmma_gemm_kernel<<<g, 256, 0, stream>>>(
        tg, Wt, ci, Wtc, bt, btc, tq,
        BP_, H_, H_, H_, H_, /*transB=*/1, /*act=*/1, 0, 0, 0);
  }
  // 8) fused block-bilinear classifier (WMMA over K=49152 on-the-fly features)
  bilinear_kernel<<<BP_ / 64, 224, 0, stream>>>(hq, tq, Wc, bc, out);
}